// GammaScorer_7842610283190
// MI455X (gfx1250) — compile-verified
//
#include <hip/hip_runtime.h>
#include <math.h>

typedef __attribute__((ext_vector_type(2))) float v2f;
typedef __attribute__((ext_vector_type(8))) float v8f;

#define DIM 128
#define EDGES_PER_WAVE 16
#define WAVES_PER_BLOCK 8

// score[e] = sigmoid( sum_d x[src[e]][d] * x[dst[e]][d] * W[d] + b )
// One wave32 handles 16 edges via 32 accumulating V_WMMA_F32_16X16X4_F32 steps.
__global__ void __launch_bounds__(WAVES_PER_BLOCK * 32)
gamma_edge_score_wmma(const float* __restrict__ x,
                      const int* __restrict__ src_idx,
                      const int* __restrict__ dst_idx,
                      const float* __restrict__ W,
                      const float* __restrict__ bias_p,
                      float* __restrict__ out,
                      int n_edges) {
    const int lane = threadIdx.x & 31;
    const int wave = (blockIdx.x * blockDim.x + threadIdx.x) >> 5;

    const int e0 = wave * EDGES_PER_WAVE;
    if (e0 >= n_edges) return;                 // wave-uniform: EXEC stays all-ones

    const int m    = lane & 15;                // edge-in-group == WMMA row M
    const int half = lane >> 4;                // K-half per ISA A/B f32 layout

    // Clamp so partial groups still gather valid rows with full EXEC
    // (WMMA requires EXEC == all ones; stores are guarded instead).
    int e = e0 + m;
    if (e >= n_edges) e = n_edges - 1;

    const float* __restrict__ rs = x + (size_t)src_idx[e] * DIM;
    const float* __restrict__ rd = x + (size_t)dst_idx[e] * DIM;

    v8f acc = {};                              // D[m][n] accumulator (16x16 f32)

#pragma unroll
    for (int k0 = 0; k0 < DIM; k0 += 4) {
        const int d = k0 + 2 * half;           // lane's K pair within this step
        // 8B gathers: exact bytes needed, row stays hot in L0 across k-loop
        const v2f s2 = *(const v2f*)(rs + d);
        const v2f d2 = *(const v2f*)(rd + d);
        const v2f w2 = *(const v2f*)(W + d);

        v2f a;                                 // A = elementwise src*dst product
        a[0] = s2[0] * d2[0];
        a[1] = s2[1] * d2[1];
        v2f bmat;                              // B = W[k] broadcast across N
        bmat[0] = w2[0];
        bmat[1] = w2[1];

        // (neg_a, A, neg_b, B, c_mod, C, reuse_a, reuse_b)
        acc = __builtin_amdgcn_wmma_f32_16x16x4_f32(
            false, a, false, bmat, (short)0, acc, false, false);
    }

    const float bias = bias_p[0];

    // D layout: VGPR r at lane l holds D[r + 8*(l>=16)][l&15].
    // Column N=0 lives in lanes 0 (rows 0..7) and 16 (rows 8..15).
    if ((lane & 15) == 0) {
        const int base = e0 + 8 * half;
#pragma unroll
        for (int r = 0; r < 8; ++r) {
            const int eo = base + r;
            if (eo < n_edges) {
                const float v = acc[r] + bias;
                out[eo] = 1.0f / (1.0f + __expf(-v));
            }
        }
    }
}

extern "C" void kernel_launch(void* const* d_in, const int* in_sizes, int n_in,
                              void* d_out, int out_size, void* d_ws, size_t ws_size,
                              hipStream_t stream) {
    const float* x       = (const float*)d_in[0];
    const int*   src_idx = (const int*)d_in[1];
    const int*   dst_idx = (const int*)d_in[2];
    const float* W       = (const float*)d_in[3];
    const float* b       = (const float*)d_in[4];
    float*       out     = (float*)d_out;

    const int n_edges = in_sizes[1];
    const int waves   = (n_edges + EDGES_PER_WAVE - 1) / EDGES_PER_WAVE;
    const int blocks  = (waves + WAVES_PER_BLOCK - 1) / WAVES_PER_BLOCK;

    hipLaunchKernelGGL(gamma_edge_score_wmma,
                       dim3(blocks), dim3(WAVES_PER_BLOCK * 32), 0, stream,
                       x, src_idx, dst_idx, W, b, out, n_edges);
}